// NumericalCayleyEmbedding_84507776516415
// MI455X (gfx1250) — compile-verified
//
#include <hip/hip_runtime.h>

// NumericalCayleyEmbedding, reduced analytically:
//   out[b,t,:] = 2*w - v   where  (I + (S - S^T) diag(x[b,t])) w = v
// One workgroup (8 wave32s) per token; 256x256 fp32 system matrix resident in
// LDS (263KB of CDNA5's 320KB WGP LDS). S is DMA'd by the Tensor Data Mover
// straight into the stride-257 padded LDS layout (TDM pad_enable), transformed
// to I + (S-S^T)diag(x) in place, then factored with blocked LU (panel=16)
// whose Schur updates run on v_wmma_f32_16x16x4_f32 with 2x2 register tiling.
// Wave index is readfirstlane'd so all tile control flow is scalar and every
// WMMA executes under full EXEC (ISA requirement).

#define N        256
#define STRIDE   257   // odd stride -> conflict-free column access in 64-bank LDS
#define NTHREADS 256
#define NWAVES   8

typedef __attribute__((ext_vector_type(2))) float        v2f;
typedef __attribute__((ext_vector_type(8))) float        v8f;
typedef __attribute__((ext_vector_type(4))) unsigned int u32x4;
typedef __attribute__((ext_vector_type(4))) int          i32x4;
typedef __attribute__((ext_vector_type(8))) int          i32x8;

#if __has_builtin(__builtin_amdgcn_tensor_load_to_lds) && \
    __has_builtin(__builtin_amdgcn_s_wait_tensorcnt)
#define HAVE_TDM 1
#endif

__global__ __launch_bounds__(NTHREADS, 1)
void cayley_rowsolve_kernel(const float* __restrict__ x,
                            const float* __restrict__ S,
                            const float* __restrict__ v,
                            float* __restrict__ out)
{
    extern __shared__ float smem[];
    float* M  = smem;                 // N x STRIDE fp32 system matrix
    float* xs = smem + N * STRIDE;    // x row
    float* vs = xs + N;               // v
    float* ws = vs + N;               // y / w (in place)

    const int tid   = threadIdx.x;
    const int lane  = tid & 31;
    // Provably wave-uniform wave index: makes the tile loops / edge guards
    // scalar (s_cbranch), so no EXEC-masked regions ever wrap a WMMA.
    const int wave  = __builtin_amdgcn_readfirstlane(tid >> 5);
    const int token = blockIdx.x;

#ifdef HAVE_TDM
    // ---- TDM: DMA S (256x256 fp32, row major) into LDS with stride-257 ----
    // pad_interval=7 (pad after every 256 DWORDs), pad_amount=0 (1 DWORD pad)
    // => each 256-float row lands at a 257-float pitch. Issued by wave 0 at
    // kernel entry so the DMA overlaps the x/v global loads below.
    if (wave == 0) {
        const unsigned long long ga = (unsigned long long)(uintptr_t)S;
#if __has_builtin(__builtin_amdgcn_groupstaticsize)
        const unsigned ldsbase = (unsigned)__builtin_amdgcn_groupstaticsize();
#else
        const unsigned ldsbase = 0u;
#endif
        u32x4 g0;
        g0[0] = 1u;                                       // count=1 valid D#
        g0[1] = ldsbase;                                  // lds_addr (bytes)
        g0[2] = (unsigned)(ga & 0xFFFFFFFFull);           // global_addr[31:0]
        g0[3] = (unsigned)((ga >> 32) & 0x01FFFFFFull)    // global_addr[56:32]
              | 0x80000000u;                              // type=2 ("image")
        i32x8 g1;
        g1[0] = (int)((2u << 16)        // data_size = 4 bytes
                    | (1u << 20)        // pad_enable
                    | (7u << 22));      // pad_interval = 256 DWORDs (pad_amount=0 -> 1 DWORD)
        g1[1] = (int)(256u << 16);      // tensor_dim0[15:0] = 256
        g1[2] = (int)(256u << 16);      // tensor_dim0 hi=0 | tensor_dim1[15:0]=256
        g1[3] = (int)(256u << 16);      // tensor_dim1 hi=0 | tile_dim0=256
        g1[4] = (int)256u;              // tile_dim1=256, tile_dim2=0
        g1[5] = (int)256u;              // tensor_dim0_stride = 256
        g1[6] = 0; g1[7] = 0;
        const i32x4 z4 = {0, 0, 0, 0};
#if __has_include(<hip/amd_detail/amd_gfx1250_TDM.h>)
        const i32x8 z8 = {0, 0, 0, 0, 0, 0, 0, 0};       // clang-23 6-arg form
        __builtin_amdgcn_tensor_load_to_lds(g0, g1, z4, z4, z8, 0);
#else
        __builtin_amdgcn_tensor_load_to_lds(g0, g1, z4, z4, 0); // ROCm7.2 5-arg
#endif
    }
#endif

    const float* xrow = x + (size_t)token * N;
    xs[tid] = xrow[tid];
    vs[tid] = v[tid];

#ifdef HAVE_TDM
    if (wave == 0) __builtin_amdgcn_s_wait_tensorcnt(0);
#else
    // Fallback: cooperative coalesced copy of S into the padded layout.
    for (int e = tid; e < N * N; e += NTHREADS) {
        const int i = e >> 8, j = e & (N - 1);
        M[i * STRIDE + j] = S[e];
    }
#endif
    __syncthreads();

    // ---- In-place: M = I + (S - S^T) * diag(x); thread t owns pair (t,j) ---
    {
        const float xi = xs[tid];
        for (int j = tid + 1; j < N; ++j) {
            const float a  = M[tid * STRIDE + j];
            const float b  = M[j * STRIDE + tid];
            const float sk = a - b;                 // skew[tid][j]
            M[tid * STRIDE + j] = sk * xs[j];       // skew[i][j]*x[j]
            M[j * STRIDE + tid] = -sk * xi;         // skew[j][i]*x[i]
        }
        M[tid * STRIDE + tid] = 1.0f;
    }
    __syncthreads();

    // ---- Blocked LU, no pivot (M = I + small-norm term, well-conditioned) --
    for (int p = 0; p < N / 16; ++p) {
        const int c0 = p * 16;
        const int tc = c0 + 16;

        // Panel factorization: columns c0..c0+15, rows c0..N-1 (unblocked).
        for (int k = 0; k < 16; ++k) {
            const int col = c0 + k;
            const float inv = 1.0f / M[col * STRIDE + col];
            const int r = col + 1 + tid;           // <=255 rows -> 1 row/thread
            if (r < N) M[r * STRIDE + col] *= inv;
            __syncthreads();
            if (r < N) {
                const float l = M[r * STRIDE + col];
                for (int j = col + 1; j < tc; ++j)
                    M[r * STRIDE + j] -= l * M[col * STRIDE + j];
            }
            __syncthreads();
        }

        // U12 = L11^{-1} A12 : each thread owns one trailing column (private
        // 16-step unit-lower forward substitution, no barriers).
        {
            const int j = tc + tid;
            if (j < N) {
                float u[16];
                u[0] = M[c0 * STRIDE + j];
                for (int k = 1; k < 16; ++k) {
                    float acc = M[(c0 + k) * STRIDE + j];
                    for (int m = 0; m < k; ++m)
                        acc -= M[(c0 + k) * STRIDE + c0 + m] * u[m];
                    u[k] = acc;
                }
                for (int k = 1; k < 16; ++k)
                    M[(c0 + k) * STRIDE + j] = u[k];
            }
        }
        __syncthreads();

        // Schur update A22 -= L21 * U12 via v_wmma_f32_16x16x4_f32.
        // 2x2 register-blocked 16x16 tiles per wave: each A/B fragment feeds
        // two WMMAs (2x LDS->WMMA density). K=16 chained as 4 K=4 WMMAs with
        // C resident in VGPRs. A negated on load (f32 WMMA has CNeg only).
        // mT/mi0/ni0/m1/n1 derive from the readfirstlane'd wave => scalar CF.
        {
            const int R  = N - tc;           // trailing dimension (mult of 16)
            const int nT = R >> 4;
            const int mT = (nT + 1) >> 1;    // 2x2 macro-tile grid
            const int ln15 = lane & 15;
            const int hi   = lane >> 4;
            for (int t = wave; t < mT * mT; t += NWAVES) {   // wave-uniform
                const int tm  = t / mT;
                const int mi0 = tm << 1;
                const int ni0 = (t - tm * mT) << 1;
                const bool m1 = (mi0 + 1) < nT;
                const bool n1 = (ni0 + 1) < nT;
                const int r0  = tc + (mi0 << 4);
                const int cc0 = tc + (ni0 << 4);
                const int cn0 = cc0 + ln15;
                const int cn1 = cn0 + 16;
                const int rb0 = r0 + (hi << 3);              // +8 rows, hi half
                const int rb1 = rb0 + 16;
                v8f c00 = {}, c01 = {}, c10 = {}, c11 = {};
#pragma unroll
                for (int rr = 0; rr < 8; ++rr) c00[rr] = M[(rb0 + rr) * STRIDE + cn0];
                if (n1) {
#pragma unroll
                    for (int rr = 0; rr < 8; ++rr) c01[rr] = M[(rb0 + rr) * STRIDE + cn1];
                }
                if (m1) {
#pragma unroll
                    for (int rr = 0; rr < 8; ++rr) c10[rr] = M[(rb1 + rr) * STRIDE + cn0];
                }
                if (m1 && n1) {
#pragma unroll
                    for (int rr = 0; rr < 8; ++rr) c11[rr] = M[(rb1 + rr) * STRIDE + cn1];
                }
#pragma unroll
                for (int kk = 0; kk < 4; ++kk) {
                    const int k0 = c0 + kk * 4;
                    const int ak = k0 + (hi << 1);  // lanes0-15 K={k0,k0+1}, hi K={k0+2,k0+3}
                    v2f a0 = {}, a1 = {}, b0 = {}, b1 = {};
                    a0.x = -M[(r0 + ln15) * STRIDE + ak];
                    a0.y = -M[(r0 + ln15) * STRIDE + ak + 1];
                    b0.x = M[ak * STRIDE + cn0];
                    b0.y = M[(ak + 1) * STRIDE + cn0];
                    if (m1) {
                        a1.x = -M[(r0 + 16 + ln15) * STRIDE + ak];
                        a1.y = -M[(r0 + 16 + ln15) * STRIDE + ak + 1];
                    }
                    if (n1) {
                        b1.x = M[ak * STRIDE + cn1];
                        b1.y = M[(ak + 1) * STRIDE + cn1];
                    }
                    c00 = __builtin_amdgcn_wmma_f32_16x16x4_f32(
                        false, a0, false, b0, (short)0, c00, false, false);
                    if (n1) c01 = __builtin_amdgcn_wmma_f32_16x16x4_f32(
                        false, a0, false, b1, (short)0, c01, false, false);
                    if (m1) c10 = __builtin_amdgcn_wmma_f32_16x16x4_f32(
                        false, a1, false, b0, (short)0, c10, false, false);
                    if (m1 && n1) c11 = __builtin_amdgcn_wmma_f32_16x16x4_f32(
                        false, a1, false, b1, (short)0, c11, false, false);
                }
#pragma unroll
                for (int rr = 0; rr < 8; ++rr) M[(rb0 + rr) * STRIDE + cn0] = c00[rr];
                if (n1) {
#pragma unroll
                    for (int rr = 0; rr < 8; ++rr) M[(rb0 + rr) * STRIDE + cn1] = c01[rr];
                }
                if (m1) {
#pragma unroll
                    for (int rr = 0; rr < 8; ++rr) M[(rb1 + rr) * STRIDE + cn0] = c10[rr];
                }
                if (m1 && n1) {
#pragma unroll
                    for (int rr = 0; rr < 8; ++rr) M[(rb1 + rr) * STRIDE + cn1] = c11[rr];
                }
            }
        }
        __syncthreads();
    }

    // ---- Triangular solves with the single RHS v ---------------------------
    ws[tid] = vs[tid];
    __syncthreads();
    // Forward: L (unit lower) y = v
    for (int k = 0; k < N - 1; ++k) {
        const float yk = ws[k];
        if (tid > k) ws[tid] -= M[tid * STRIDE + k] * yk;
        __syncthreads();
    }
    // Backward: U w = y
    for (int k = N - 1; k >= 0; --k) {
        if (tid == k) ws[k] /= M[k * STRIDE + k];
        __syncthreads();
        const float wk = ws[k];
        if (tid < k) ws[tid] -= M[tid * STRIDE + k] * wk;
        __syncthreads();
    }

    // ---- out = 2w - v ------------------------------------------------------
    out[(size_t)token * N + tid] = 2.0f * ws[tid] - vs[tid];
}

extern "C" void kernel_launch(void* const* d_in, const int* in_sizes, int n_in,
                              void* d_out, int out_size, void* d_ws, size_t ws_size,
                              hipStream_t stream) {
    const float* x = (const float*)d_in[0];   // (B, T, 256) fp32
    const float* S = (const float*)d_in[1];   // (256, 256) fp32
    const float* v = (const float*)d_in[2];   // (1, 256) fp32
    float* out = (float*)d_out;               // (B, T, 256) fp32

    const int tokens = in_sizes[0] / N;       // B*T = 1024
    const size_t lds_bytes = (size_t)(N * STRIDE + 3 * N) * sizeof(float); // ~266 KB

    // CDNA5 WGP has 320KB LDS; raise the dynamic-LDS cap (deterministic,
    // graph-capture safe: no alloc/sync/stream ops).
    (void)hipFuncSetAttribute((const void*)cayley_rowsolve_kernel,
                              hipFuncAttributeMaxDynamicSharedMemorySize,
                              (int)lds_bytes);

    cayley_rowsolve_kernel<<<dim3(tokens), dim3(NTHREADS), lds_bytes, stream>>>(
        x, S, v, out);
}